// GAT_77704548319854
// MI455X (gfx1250) — compile-verified
//
#include <hip/hip_runtime.h>
#include <hip/hip_bf16.h>

typedef __attribute__((ext_vector_type(2))) float v2f;
typedef __attribute__((ext_vector_type(8))) float v8f;

#define WAVES 8      // waves per block (256 threads)
#define LDSTR 68     // padded LDS row stride in floats: 68*4B = 272B -> 16B aligned rows,
                     // stride mod 64 banks = 4 -> conflict-free column access

// ---- kernel 1: zero the per-node source flags -------------------------------
__global__ void gat_zero_flags(float* __restrict__ f, int n) {
  int i = blockIdx.x * blockDim.x + threadIdx.x;
  if (i < n) f[i] = 0.0f;
}

// ---- kernel 2: mark nodes that appear as an edge source ---------------------
// (all writers store the same 1.0f -> race is benign and deterministic)
__global__ void gat_mark_src(const int* __restrict__ trip, float* __restrict__ f, int e) {
  int i = blockIdx.x * blockDim.x + threadIdx.x;
  if (i < e) f[trip[3 * i]] = 1.0f;
}

// ---- kernel 3: out[v,:] = flag[v] * relu( clip(emb[v,:]) @ W_a^T + b_a ) ----
__global__ __launch_bounds__(256) void gat_node_wmma(
    const float* __restrict__ emb,   // N x 64
    const float* __restrict__ Wa,    // 64 x 64 row-major (OUT x IN)
    const float* __restrict__ ba,    // 64
    const float* __restrict__ flags, // N
    float* __restrict__ out,         // N x 64
    int n_nodes)
{
  __shared__ float lw[64 * LDSTR];          // W_a staged, padded rows
  __shared__ float ht[WAVES][16 * LDSTR];   // per-wave 16x64 embedding tile
  __shared__ float scl[WAVES][16];          // per-row clip scale
  __shared__ float flg[WAVES][16];          // per-row source flag

  const int tid = threadIdx.x;
  const int w   = tid >> 5;
  const int L   = tid & 31;

  // Stage W_a (64x64 = 4096 floats) with all 256 threads: 16 floats each.
  {
    int n  = tid >> 2;
    int c0 = (tid & 3) * 16;
    const float4* s4 = (const float4*)(Wa + n * 64 + c0);
    float4* d4 = (float4*)&lw[n * LDSTR + c0];
#pragma unroll
    for (int j = 0; j < 4; ++j) d4[j] = s4[j];
  }

  const int tile   = blockIdx.x * WAVES + w;
  const int r0     = tile * 16;
  const bool active = (r0 < n_nodes);

  if (active) {
    // Each lane stages half a row (32 floats) and accumulates its sum of squares.
    int m  = L >> 1;
    int c0 = (L & 1) * 32;
    int row = r0 + m;
    if (row >= n_nodes) row = n_nodes - 1;   // clamp (N is a multiple of 16 here)
    const float4* g4 = (const float4*)(emb + (size_t)row * 64 + c0);
    float4* h4 = (float4*)&ht[w][m * LDSTR + c0];
    float ss = 0.0f;
#pragma unroll
    for (int j = 0; j < 8; ++j) {
      float4 v = g4[j];
      ss += v.x * v.x + v.y * v.y + v.z * v.z + v.w * v.w;
      h4[j] = v;
    }
    ss += __shfl_xor(ss, 1, 32);             // combine the two half-rows
    if ((L & 1) == 0) {
      float nrm = sqrtf(ss);
      scl[w][m] = fminf(1.0f, 1.0f / (nrm + 1e-7f));
    }
    if (L < 16) {
      int fr = r0 + L;
      flg[w][L] = (fr < n_nodes) ? flags[fr] : 0.0f;
    }
  }
  __syncthreads();

  if (active) {
    v8f acc[4] = {v8f{}, v8f{}, v8f{}, v8f{}};

    const int mm  = L & 15;          // A: row M, B: col N (within 16-tile)
    const int khi = (L >> 4) * 2;    // high half-wave holds K+2 / K+3
    const float sc = scl[w][mm];
    const float* arow = &ht[w][mm * LDSTR];

#pragma unroll
    for (int kk = 0; kk < 64; kk += 4) {
      const int kb = kk + khi;
      v2f a;
      a.x = arow[kb]     * sc;
      a.y = arow[kb + 1] * sc;
#pragma unroll
      for (int nt = 0; nt < 4; ++nt) {
        const float* brow = &lw[(nt * 16 + mm) * LDSTR + kb];
        v2f b;
        b.x = brow[0];
        b.y = brow[1];
        acc[nt] = __builtin_amdgcn_wmma_f32_16x16x4_f32(
            /*neg_a=*/false, a, /*neg_b=*/false, b,
            /*c_mod=*/(short)0, acc[nt],
            /*reuse_a=*/false, /*reuse_b=*/false);
      }
    }

    // C/D layout: VGPR j, lanes 0-15 -> row j, lanes 16-31 -> row j+8; col = lane&15.
    const int hi8 = (L >> 4) * 8;
    float fr[8];
#pragma unroll
    for (int j = 0; j < 8; ++j) fr[j] = flg[w][j + hi8];

#pragma unroll
    for (int nt = 0; nt < 4; ++nt) {
      const float bias = ba[nt * 16 + mm];
#pragma unroll
      for (int j = 0; j < 8; ++j) {
        int row = r0 + j + hi8;
        if (row < n_nodes) {
          float v = fmaxf(acc[nt][j] + bias, 0.0f) * fr[j];
          out[(size_t)row * 64 + nt * 16 + mm] = v;
        }
      }
    }
  }
}

extern "C" void kernel_launch(void* const* d_in, const int* in_sizes, int n_in,
                              void* d_out, int out_size, void* d_ws, size_t ws_size,
                              hipStream_t stream) {
  const int*   trip = (const int*)d_in[0];     // E x 3
  const float* emb  = (const float*)d_in[1];   // N x 64
  const float* Wa   = (const float*)d_in[2];   // 64 x 64
  const float* ba   = (const float*)d_in[3];   // 64
  // d_in[4] = W_a2, d_in[5] = b_a2: algebraically eliminated (see analysis)
  float* out   = (float*)d_out;
  float* flags = (float*)d_ws;                 // N floats of scratch

  const int E = in_sizes[0] / 3;
  const int N = in_sizes[1] / 64;

  gat_zero_flags<<<(N + 255) / 256, 256, 0, stream>>>(flags, N);
  gat_mark_src<<<(E + 255) / 256, 256, 0, stream>>>(trip, flags, E);

  const int tiles  = (N + 15) / 16;
  const int blocks = (tiles + WAVES - 1) / WAVES;
  gat_node_wmma<<<blocks, 256, 0, stream>>>(emb, Wa, ba, flags, out, N);
}